// DeformableSpatialAttention_90288802496931
// MI455X (gfx1250) — compile-verified
//
#include <hip/hip_runtime.h>
#include <cmath>

// ---------------------------------------------------------------------------
// DeformableSpatialAttention on gfx1250 (CDNA5, wave32)
// C=256, Hh=8, P=12, L=4, H=W=64, bs=6, nq=4096, dph=32 (4 ch/head)
//
// Working set is L2-resident (192MB); binding resource is L2 bandwidth.
// GEMM blocks: 8 waves = 8 M-tiles, each wave computes NT 16x16 C tiles from
// one shared A fragment (NT independent accumulator chains hide WMMA latency,
// A re-reads cut ~2.5x). B panel staged in LDS in k-pair-interleaved layout so
// each WMMA B fragment is a single aligned ds_load_b64 (no register re-pairing
// movs); row stride chosen == 32 (mod 64) dwords -> bank-conflict-free across
// the two half-wave k-pair rows. Exact-f32 math via V_WMMA_F32_16X16X4_F32.
// ---------------------------------------------------------------------------

typedef float v2f __attribute__((ext_vector_type(2)));
typedef float v8f __attribute__((ext_vector_type(8)));

#define HWQ   4096              // h*w = 64*64
#define CDIM  256
#define BS    6                 // bs_org * nframes
#define MQ    (BS * HWQ)        // 24576 rows

// ---- input layout change: [2,256,3,64,64] -> [6,4096,256] -----------------
__global__ __launch_bounds__(256) void flatten_kernel(
    const float* __restrict__ fea, float* __restrict__ q, int total) {
  int tid = blockIdx.x * blockDim.x + threadIdx.x;
  if (tid >= total) return;
  int c  = tid & 255;
  int p  = (tid >> 8) & 4095;
  int bf = tid >> 20;                 // 4096*256 = 2^20
  int b = bf / 3, f = bf - 3 * b;
  q[tid] = fea[((size_t)(b * 256 + c) * 3 + f) * 4096 + p];
}

// ---- learned positional encoding: qpos[y*64+x][c] --------------------------
__global__ __launch_bounds__(256) void qpos_kernel(
    const float* __restrict__ col, const float* __restrict__ row,
    float* __restrict__ qpos, int total) {
  int tid = blockIdx.x * blockDim.x + threadIdx.x;
  if (tid >= total) return;
  int c = tid & 255;
  int q = tid >> 8;
  int x = q & 63, y = q >> 6;
  qpos[tid] = (c < 128) ? col[x * 128 + c] : row[y * 128 + (c - 128)];
}

// ---- pack [Woff | Wattn] -> Wp[256,288], [boff | battn] -> bp[288] ---------
__global__ __launch_bounds__(256) void pack_proj_kernel(
    const float* __restrict__ Woff, const float* __restrict__ Wattn,
    const float* __restrict__ boff, const float* __restrict__ battn,
    float* __restrict__ Wp, float* __restrict__ bp) {
  int t = blockIdx.x * blockDim.x + threadIdx.x;
  if (t < 288) bp[t] = (t < 192) ? boff[t] : battn[t - 192];
  if (t >= 256 * 288) return;
  int r = t / 288, c = t - r * 288;
  Wp[t] = (c < 192) ? Woff[r * 192 + c] : Wattn[r * 96 + (c - 192)];
}

// ---- GEMM: C[M,N] = (A [+qpos]) * B + bias (+ addend) ----------------------
// A row-major [M,lda]; B row-major [K,LDB]. 32-bit WMMA fragment layout
// (ISA 7.12.2): lanes 0-15 hold K=k,k+1; lanes 16-31 hold K=k+2,k+3.
// LDS B layout: element B[2*rp+e][col] at Bs[rp*RS + 2*col + e], so a wave's
// B fragment {B[ka][n], B[ka+1][n]} is one contiguous 8B ds_load.
template <int K, int LDB, int NT, bool HAS_POS, bool HAS_ADD>
__global__ __launch_bounds__(256) void gemm_wmma_f32(
    const float* __restrict__ A, int lda,
    const float* __restrict__ qpos,               // used iff HAS_POS
    const float* __restrict__ B,
    const float* __restrict__ bias,
    const float* __restrict__ addend, int ldadd,  // used iff HAS_ADD
    float* __restrict__ C, int ldc) {
  constexpr int KC = (K < 64) ? K : 64;           // K-chunk staged per barrier
  constexpr int NW = 16 * NT;                     // N width per block
  // Row stride (dwords) between k-pair rows: keep RS % 64 == 32 so the two
  // half-wave rows land on disjoint bank halves.
  constexpr int RS = ((2 * NW) % 64 == 32) ? (2 * NW) : (2 * NW + 32);
  constexpr int UNITS = (KC / 2) * (NW / 2);      // (row-pair, col-pair) units
  static_assert(UNITS % 256 == 0, "staging must tile exactly");
  __shared__ __align__(16) float Bs[(KC / 2) * RS];

  const int wave = threadIdx.x >> 5;              // uniform within wave
  const int lane = threadIdx.x & 31;
  const int half = lane >> 4;                     // 0: K{0,1}, 1: K{+2,+3}
  const int l15  = lane & 15;
  const int tm = (blockIdx.y * 8 + wave) << 4;    // per-wave M tile
  const int tn = blockIdx.x * NW;                 // block's N range
  const int arow = tm + l15;

  const float* __restrict__ Arow = A + (size_t)arow * lda;
  const float* __restrict__ Qrow =
      HAS_POS ? (qpos + (size_t)(arow & (HWQ - 1)) * lda) : nullptr;

  v8f acc[NT];
  #pragma unroll
  for (int j = 0; j < NT; ++j) acc[j] = {};

  for (int kc = 0; kc < K; kc += KC) {
    __syncthreads();
    #pragma unroll
    for (int uu = 0; uu < UNITS; uu += 256) {
      const int u   = uu + threadIdx.x;
      const int rp  = u / (NW / 2);               // k-pair row
      const int col = (u - rp * (NW / 2)) << 1;   // even column
      const float2 g0 = *reinterpret_cast<const float2*>(
          B + (size_t)(kc + 2 * rp) * LDB + tn + col);
      const float2 g1 = *reinterpret_cast<const float2*>(
          B + (size_t)(kc + 2 * rp + 1) * LDB + tn + col);
      float4 w; w.x = g0.x; w.y = g1.x; w.z = g0.y; w.w = g1.y;
      *reinterpret_cast<float4*>(&Bs[rp * RS + 2 * col]) = w;
    }
    __syncthreads();                              // EXEC all-ones past here

    #pragma unroll
    for (int kk = 0; kk < KC; kk += 4) {
      const int ka = kk + (half << 1);            // even
      v2f a;
      a.x = Arow[kc + ka];
      a.y = Arow[kc + ka + 1];
      if constexpr (HAS_POS) {
        a.x += Qrow[kc + ka];
        a.y += Qrow[kc + ka + 1];
      }
      #pragma unroll
      for (int j = 0; j < NT; ++j) {              // NT independent acc chains
        const v2f b = *reinterpret_cast<const v2f*>(
            &Bs[(ka >> 1) * RS + 2 * (j * 16 + l15)]);
        acc[j] = __builtin_amdgcn_wmma_f32_16x16x4_f32(
            /*neg_a=*/false, a, /*neg_b=*/false, b,
            /*c_mod=*/(short)0, acc[j], /*reuse_a=*/false, /*reuse_b=*/false);
      }
    }
  }

  #pragma unroll
  for (int j = 0; j < NT; ++j) {
    const int col = tn + j * 16 + l15;
    const float bv = bias[col];
    #pragma unroll
    for (int r = 0; r < 8; ++r) {
      const int row = tm + r + (half << 3);       // lanes 16-31 hold rows +8
      float v = acc[j][r] + bv;
      if constexpr (HAS_ADD) v += addend[(size_t)row * ldadd + col];
      C[(size_t)row * ldc + col] = v;
    }
  }
}

// ---- softmax + bilinear deformable gather ---------------------------------
// proj[b,q,0:192] = offsets (h*24+p*2+d), proj[b,q,192:288] = attn logits.
// val[b, pix, h*4 + c]: 4 channels per head, 16B contiguous -> float4 loads.
// gx = x + off_x, gy = y + off_y (algebraic simplification of reference).
__global__ __launch_bounds__(256) void gather_kernel(
    const float* __restrict__ proj, const float* __restrict__ val,
    float* __restrict__ out32) {
  int tid = blockIdx.x * blockDim.x + threadIdx.x;
  if (tid >= MQ * 8) return;
  const int h = tid & 7;
  const int q = (tid >> 3) & 4095;
  const int b = tid >> 15;

  const float* __restrict__ pr = proj + ((size_t)b * HWQ + q) * 288;
  const float fx = (float)(q & 63);
  const float fy = (float)(q >> 6);

  float gx[12], gy[12], lg[12];
  float mx = -3.4e38f;
  #pragma unroll
  for (int p = 0; p < 12; ++p) {
    gx[p] = fx + pr[h * 24 + p * 2];
    gy[p] = fy + pr[h * 24 + p * 2 + 1];
    lg[p] = pr[192 + h * 12 + p];
    mx = fmaxf(mx, lg[p]);
  }
  float s = 0.f;
  #pragma unroll
  for (int p = 0; p < 12; ++p) { lg[p] = __expf(lg[p] - mx); s += lg[p]; }
  const float inv = 1.f / s;

  float a0 = 0.f, a1 = 0.f, a2 = 0.f, a3 = 0.f;
  const float* __restrict__ vb = val + (size_t)b * HWQ * 32 + h * 4;

  #pragma unroll
  for (int p = 0; p < 12; ++p) {
    const float w = lg[p] * inv;
    const float x0f = floorf(gx[p]), y0f = floorf(gy[p]);
    const float lx = gx[p] - x0f, ly = gy[p] - y0f;
    const int x0 = (int)x0f, y0 = (int)y0f;
    const float cw[4] = {(1.f - lx) * (1.f - ly), lx * (1.f - ly),
                         (1.f - lx) * ly,         lx * ly};
    const int cx[4] = {x0, x0 + 1, x0, x0 + 1};
    const int cy[4] = {y0, y0, y0 + 1, y0 + 1};
    #pragma unroll
    for (int cc = 0; cc < 4; ++cc) {
      const int xi = cx[cc], yi = cy[cc];
      if (xi >= 0 && xi < 64 && yi >= 0 && yi < 64) {
        const float4 vv =
            *reinterpret_cast<const float4*>(vb + (size_t)(yi * 64 + xi) * 32);
        const float ww = w * cw[cc];
        a0 += ww * vv.x; a1 += ww * vv.y; a2 += ww * vv.z; a3 += ww * vv.w;
      }
    }
  }
  float4 o; o.x = a0; o.y = a1; o.z = a2; o.w = a3;
  *reinterpret_cast<float4*>(out32 + ((size_t)b * HWQ + q) * 32 + h * 4) = o;
}

// ---------------------------------------------------------------------------
extern "C" void kernel_launch(void* const* d_in, const int* in_sizes, int n_in,
                              void* d_out, int out_size, void* d_ws, size_t ws_size,
                              hipStream_t stream) {
  (void)in_sizes; (void)n_in; (void)out_size; (void)ws_size;
  const float* rgb_fea = (const float*)d_in[0];
  const float* ir_fea  = (const float*)d_in[1];
  const float* pcol[2] = {(const float*)d_in[2], (const float*)d_in[4]};
  const float* prow[2] = {(const float*)d_in[3], (const float*)d_in[5]};
  const float* Woff[2]  = {(const float*)d_in[6],  (const float*)d_in[14]};
  const float* boff[2]  = {(const float*)d_in[7],  (const float*)d_in[15]};
  const float* Wattn[2] = {(const float*)d_in[8],  (const float*)d_in[16]};
  const float* battn[2] = {(const float*)d_in[9],  (const float*)d_in[17]};
  const float* Wval[2]  = {(const float*)d_in[10], (const float*)d_in[18]};
  const float* bval[2]  = {(const float*)d_in[11], (const float*)d_in[19]};
  const float* Wout[2]  = {(const float*)d_in[12], (const float*)d_in[20]};
  const float* bout[2]  = {(const float*)d_in[13], (const float*)d_in[21]};

  float* ws = (float*)d_ws;
  size_t woff = 0;
  auto alloc = [&](size_t n) {
    float* p = ws + woff;
    woff += (n + 63) & ~(size_t)63;   // 256B alignment for float4 paths
    return p;
  };
  float* qb[2][2];                     // [mod][pingpong]
  qb[0][0] = alloc((size_t)MQ * 256);
  qb[0][1] = alloc((size_t)MQ * 256);
  qb[1][0] = alloc((size_t)MQ * 256);
  qb[1][1] = alloc((size_t)MQ * 256);
  float* qpos[2] = {alloc((size_t)HWQ * 256), alloc((size_t)HWQ * 256)};
  float* valbuf  = alloc((size_t)MQ * 32);
  float* projbuf = alloc((size_t)MQ * 288);
  float* o32     = alloc((size_t)MQ * 32);
  float* Wproj[2][4];
  float* bproj[2][4];
  for (int m = 0; m < 2; ++m)
    for (int l = 0; l < 4; ++l) {
      Wproj[m][l] = alloc((size_t)256 * 288);
      bproj[m][l] = alloc(288);
    }

  // ---- input prep ----
  {
    const int total = MQ * 256;
    const int blocks = (total + 255) / 256;
    flatten_kernel<<<blocks, 256, 0, stream>>>(rgb_fea, qb[0][0], total);
    flatten_kernel<<<blocks, 256, 0, stream>>>(ir_fea,  qb[1][0], total);
  }
  {
    const int total = HWQ * 256;
    const int blocks = (total + 255) / 256;
    qpos_kernel<<<blocks, 256, 0, stream>>>(pcol[0], prow[0], qpos[0], total);
    qpos_kernel<<<blocks, 256, 0, stream>>>(pcol[1], prow[1], qpos[1], total);
  }
  {
    const int blocks = (256 * 288 + 255) / 256;
    for (int m = 0; m < 2; ++m)
      for (int l = 0; l < 4; ++l)
        pack_proj_kernel<<<blocks, 256, 0, stream>>>(
            Woff[m] + (size_t)l * 256 * 192, Wattn[m] + (size_t)l * 256 * 96,
            boff[m] + (size_t)l * 192,       battn[m] + (size_t)l * 96,
            Wproj[m][l], bproj[m][l]);
  }

  // ---- layers ----
  int cur = 0;
  for (int l = 0; l < 4; ++l) {
    const int nxt = cur ^ 1;
    for (int mod = 0; mod < 2; ++mod) {
      const float* qs = qb[mod][cur];        // query of this modality
      const float* qo = qb[mod ^ 1][cur];    // value source = other modality
      const float* Wv = Wval[mod] + (size_t)l * 256 * 32;
      const float* bv = bval[mod] + (size_t)l * 32;
      const float* Wu = Wout[mod] + (size_t)l * 32 * 256;
      const float* bu = bout[mod] + (size_t)l * 256;

      // 1. value = q_other @ Wval + bval           [MQ,32]  (NT=2 -> 1 pass)
      gemm_wmma_f32<256, 32, 2, false, false>
          <<<dim3(1, MQ / 128), 256, 0, stream>>>(
              qo, 256, nullptr, Wv, bv, nullptr, 0, valbuf, 32);
      // 2. proj = (q+qpos) @ [Woff|Wattn] + bias   [MQ,288] (NT=3 -> 6 passes)
      gemm_wmma_f32<256, 288, 3, true, false>
          <<<dim3(288 / 48, MQ / 128), 256, 0, stream>>>(
              qs, 256, qpos[mod], Wproj[mod][l], bproj[mod][l],
              nullptr, 0, projbuf, 288);
      // 3. softmax + deformable bilinear gather -> out32  [MQ,32]
      {
        const int total = MQ * 8;
        gather_kernel<<<(total + 255) / 256, 256, 0, stream>>>(projbuf, valbuf, o32);
      }
      // 4. q_new = out32 @ Wout + bout + q  (last layer -> d_out directly)
      float* dst = (l == 3) ? ((float*)d_out + (size_t)mod * MQ * 256)
                            : qb[mod][nxt];
      gemm_wmma_f32<32, 256, 4, false, true>
          <<<dim3(256 / 64, MQ / 128), 256, 0, stream>>>(
              o32, 32, nullptr, Wu, bu, qs, 256, dst, 256);
    }
    cur ^= 1;
  }
}